// GNN_cell_67877663146718
// MI455X (gfx1250) — compile-verified
//
#include <hip/hip_runtime.h>
#include <math.h>

#define DIM 64

typedef __attribute__((ext_vector_type(2))) float v2f;
typedef __attribute__((ext_vector_type(8))) float v8f;

__device__ __forceinline__ float selu_f(float x) {
  const float sc = 1.0507009873554805f;
  const float al = 1.6732632423543772f;
  return x > 0.f ? sc * x : sc * al * (expf(x) - 1.f);
}

// float atomic max via monotonic integer mapping (handles mixed signs, -inf init)
__device__ __forceinline__ void atomicMaxF(float* addr, float v) {
  if (v >= 0.f) atomicMax((int*)addr, __float_as_int(v));
  else          atomicMin((unsigned int*)addr, __float_as_uint(v));
}

// ---------------------------------------------------------------------------
// WMMA GEMM: Y[M,64] = X[M,K] @ W[K,64], K in {3,64}, M % 128 == 0.
// Block = 256 threads (8 waves); each wave computes a 16x64 tile with
// 4 f32 16x16x4 accumulators; k-loop fully unrolled (K is a template arg).
// W is staged TRANSPOSED in LDS so each (k,k+1) pair is one aligned b64 read.
// A layout (32-bit 16x4): lanes 0-15 -> K={k0,k0+1}; lanes 16-31 -> {k0+2,k0+3}
// C/D layout: VGPR i -> M=i (lanes 0-15), M=i+8 (lanes 16-31)
// ---------------------------------------------------------------------------
template <int K, int EPI>
__global__ void gemm_n64(const float* __restrict__ X, const float* __restrict__ W,
                         float* __restrict__ Y) {
  __shared__ float LW[DIM * DIM];   // LW[col*64 + k] = W[k*64 + col]
  for (int t = threadIdx.x; t < K * DIM; t += 256) {
    const int k = t >> 6, c = t & 63;
    LW[c * DIM + k] = W[t];
  }
  __syncthreads();

  const int wave = threadIdx.x >> 5;
  const int lane = threadIdx.x & 31;
  const int row0 = blockIdx.x * 128 + wave * 16;
  const int lr = lane & 15;
  const int hi = lane >> 4;
  const int r  = row0 + lr;

  float accv[4][8];
#if defined(__gfx1250__)
  const v8f zero = {0.f, 0.f, 0.f, 0.f, 0.f, 0.f, 0.f, 0.f};
  v8f acc[4];
  acc[0] = zero; acc[1] = zero; acc[2] = zero; acc[3] = zero;

  if (K == 3) {
    // single padded k-step; unconditional loads + lane selects (no divergence)
    const float x0 = X[(size_t)r * 3 + 0];
    const float x1 = X[(size_t)r * 3 + 1];
    const float x2 = X[(size_t)r * 3 + 2];
    v2f a;
    a.x = hi ? x2 : x0;
    a.y = hi ? 0.f : x1;
#pragma unroll
    for (int nt = 0; nt < 4; ++nt) {
      const int col = nt * 16 + lr;
      const float w0 = LW[col * DIM + 0];
      const float w1 = LW[col * DIM + 1];
      const float w2 = LW[col * DIM + 2];
      v2f b;
      b.x = hi ? w2 : w0;
      b.y = hi ? 0.f : w1;
      acc[nt] = __builtin_amdgcn_wmma_f32_16x16x4_f32(
          false, a, false, b, (short)0, acc[nt], false, false);
    }
  } else {
    const float* xrow = X + (size_t)r * K;
#pragma unroll
    for (int k0 = 0; k0 < K; k0 += 4) {
      const int ka = k0 + 2 * hi;
      const v2f a = *(const v2f*)(xrow + ka);
#pragma unroll
      for (int nt = 0; nt < 4; ++nt) {
        const int col = nt * 16 + lr;
        const v2f b = *(const v2f*)(&LW[col * DIM + ka]);
        acc[nt] = __builtin_amdgcn_wmma_f32_16x16x4_f32(
            false, a, false, b, (short)0, acc[nt], false, false);
      }
    }
  }
#pragma unroll
  for (int nt = 0; nt < 4; ++nt)
#pragma unroll
    for (int i = 0; i < 8; ++i) accv[nt][i] = acc[nt][i];
#else
  // host-pass / non-gfx1250 placeholder (never executed on target)
  for (int nt = 0; nt < 4; ++nt)
    for (int i = 0; i < 8; ++i) accv[nt][i] = 0.f;
#endif

#pragma unroll
  for (int nt = 0; nt < 4; ++nt) {
#pragma unroll
    for (int i = 0; i < 8; ++i) {
      float vv = accv[nt][i];
      if (EPI == 1) vv = selu_f(vv);
      Y[(size_t)(row0 + 8 * hi + i) * DIM + nt * 16 + lr] = vv;
    }
  }
}

// per-node attention scores: SS = h . a_src, SD = h . a_dst
__global__ void scores_k(const float* __restrict__ H, const float* __restrict__ as,
                         const float* __restrict__ ad, float* __restrict__ SS,
                         float* __restrict__ SD, int n) {
  int i = blockIdx.x * blockDim.x + threadIdx.x;
  if (i >= n) return;
  const float* h = H + (size_t)i * DIM;
  float s0 = 0.f, s1 = 0.f;
#pragma unroll 8
  for (int c = 0; c < DIM; ++c) { float v = h[c]; s0 += v * as[c]; s1 += v * ad[c]; }
  SS[i] = s0; SD[i] = s1;
}

__global__ void init_agg_k(float* __restrict__ AGG, float* __restrict__ MX,
                           float* __restrict__ Z, int n) {
  int i = blockIdx.x * blockDim.x + threadIdx.x;
  if (i < n * DIM) AGG[i] = 0.f;
  if (i < n) { MX[i] = -__builtin_inff(); Z[i] = 0.f; }
}

__global__ void edge1_k(const int* __restrict__ src, const int* __restrict__ dst,
                        const float* __restrict__ SS, const float* __restrict__ SD,
                        float* __restrict__ P, float* __restrict__ MX, int E) {
  int e = blockIdx.x * blockDim.x + threadIdx.x;
  if (e >= E) return;
  float s = SS[src[e]] + SD[dst[e]];
  float l = s >= 0.f ? s : 0.2f * s;   // leaky_relu(., 0.2)
  P[e] = l;
  atomicMaxF(&MX[dst[e]], l);
}

__global__ void edge2_k(const int* __restrict__ dst, float* __restrict__ P,
                        const float* __restrict__ MX, float* __restrict__ Z, int E) {
  int e = blockIdx.x * blockDim.x + threadIdx.x;
  if (e >= E) return;
  int d = dst[e];
  float p = expf(P[e] - MX[d]);
  P[e] = p;
  atomicAdd(&Z[d], p);
}

// weighted scatter-add: AGG[dst] += (p/z[dst]) * H[src], 4 channels per thread
__global__ void edge3_k(const int* __restrict__ src, const int* __restrict__ dst,
                        const float* __restrict__ P, const float* __restrict__ Z,
                        const float* __restrict__ H, float* __restrict__ AGG, int E) {
  int t = blockIdx.x * blockDim.x + threadIdx.x;
  int e = t >> 4;
  if (e >= E) return;
  int g = (t & 15) * 4;
  int d = dst[e], s = src[e];
  float a = P[e] / Z[d];
  const float4 hv = *(const float4*)(H + (size_t)s * DIM + g);
  float* out = AGG + (size_t)d * DIM + g;
  atomicAdd(out + 0, a * hv.x);
  atomicAdd(out + 1, a * hv.y);
  atomicAdd(out + 2, a * hv.z);
  atomicAdd(out + 3, a * hv.w);
}

__global__ void fill_k(float* __restrict__ p, float v, int n) {
  int i = blockIdx.x * blockDim.x + threadIdx.x;
  if (i < n) p[i] = v;
}

// relu(agg + bias) then segment_max into POOL (values >= 0 -> int atomicMax, 0 init)
__global__ void pool_k(const float* __restrict__ AGG, const float* __restrict__ bias,
                       const int* __restrict__ cl, float* __restrict__ POOL, int n_in) {
  int t = blockIdx.x * blockDim.x + threadIdx.x;
  if (t >= n_in * DIM) return;
  int n = t >> 6, c = t & 63;
  float v = AGG[t] + bias[c];
  v = v > 0.f ? v : 0.f;
  atomicMax((int*)&POOL[(size_t)cl[n] * DIM + c], __float_as_int(v));
}

__global__ void bn_stats_k(const float* __restrict__ X, float* __restrict__ stat, int n) {
  int c = blockIdx.x;   // 0..63
  float s = 0.f, q = 0.f;
  for (int i = threadIdx.x; i < n; i += blockDim.x) {
    float v = X[(size_t)i * DIM + c];
    s += v; q += v * v;
  }
  __shared__ float sh[256], qh[256];
  sh[threadIdx.x] = s; qh[threadIdx.x] = q;
  __syncthreads();
  for (int o = blockDim.x / 2; o > 0; o >>= 1) {
    if ((int)threadIdx.x < o) { sh[threadIdx.x] += sh[threadIdx.x + o]; qh[threadIdx.x] += qh[threadIdx.x + o]; }
    __syncthreads();
  }
  if (threadIdx.x == 0) {
    float mu = sh[0] / n;
    stat[2 * c]     = mu;
    stat[2 * c + 1] = qh[0] / n - mu * mu;
  }
}

__global__ void bn_apply_k(const float* __restrict__ X, const float* __restrict__ stat,
                           float* __restrict__ Y, int n) {
  int t = blockIdx.x * blockDim.x + threadIdx.x;
  if (t >= n * DIM) return;
  int c = t & 63;
  Y[t] = (X[t] - stat[2 * c]) * rsqrtf(stat[2 * c + 1] + 1e-5f);
}

// Bm[m,v] = ALL[v,m,:] . w_attn
__global__ void bm_k(const float* __restrict__ ALLV, const float* __restrict__ wa,
                     float* __restrict__ BM, int M) {
  int m = blockIdx.x * blockDim.x + threadIdx.x;
  if (m >= M) return;
  for (int v = 0; v < 4; ++v) {
    const float* h = ALLV + ((size_t)v * M + m) * DIM;
    float s = 0.f;
#pragma unroll 8
    for (int c = 0; c < DIM; ++c) s += h[c] * wa[c];
    BM[m * 4 + v] = s;
  }
}

__global__ void gram16_k(const float* __restrict__ BM, float* __restrict__ G, int M) {
  float acc[16];
#pragma unroll
  for (int k = 0; k < 16; ++k) acc[k] = 0.f;
  for (int m = threadIdx.x; m < M; m += blockDim.x) {
    float b[4] = {BM[m * 4], BM[m * 4 + 1], BM[m * 4 + 2], BM[m * 4 + 3]};
#pragma unroll
    for (int i = 0; i < 4; ++i)
#pragma unroll
      for (int j = 0; j < 4; ++j) acc[i * 4 + j] += b[i] * b[j];
  }
  __shared__ float sh[256];
  for (int k = 0; k < 16; ++k) {
    sh[threadIdx.x] = acc[k];
    __syncthreads();
    for (int o = 128; o > 0; o >>= 1) {
      if ((int)threadIdx.x < o) sh[threadIdx.x] += sh[threadIdx.x + o];
      __syncthreads();
    }
    if (threadIdx.x == 0) G[k] = sh[0];
    __syncthreads();
  }
}

// A = row-softmax(leaky_relu(G[i][j]/(|bi||bj|), 0.1))
__global__ void attn4_k(const float* __restrict__ G, float* __restrict__ A) {
  if (threadIdx.x != 0 || blockIdx.x != 0) return;
  float nrm[4];
  for (int v = 0; v < 4; ++v) nrm[v] = sqrtf(G[v * 4 + v]);
  for (int i = 0; i < 4; ++i) {
    float row[4]; float mx = -__builtin_inff();
    for (int j = 0; j < 4; ++j) {
      float g = G[i * 4 + j] / (nrm[i] * nrm[j]);
      g = g >= 0.f ? g : 0.1f * g;
      row[j] = g; mx = fmaxf(mx, g);
    }
    float s = 0.f;
    for (int j = 0; j < 4; ++j) { row[j] = expf(row[j] - mx); s += row[j]; }
    for (int j = 0; j < 4; ++j) A[i * 4 + j] = row[j] / s;
  }
}

// ALLF[l,m,c] = sum_k ALL[k,m,c] * A[k,l]
__global__ void fuse_k(const float* __restrict__ ALLV, const float* __restrict__ A,
                       float* __restrict__ ALLF, int M) {
  int t = blockIdx.x * blockDim.x + threadIdx.x;
  if (t >= M * DIM) return;
  size_t MD = (size_t)M * DIM;
  float x0 = ALLV[t], x1 = ALLV[MD + t], x2 = ALLV[2 * MD + t], x3 = ALLV[3 * MD + t];
#pragma unroll
  for (int l = 0; l < 4; ++l)
    ALLF[l * MD + t] = x0 * A[l] + x1 * A[4 + l] + x2 * A[8 + l] + x3 * A[12 + l];
}

// mu[c] = mean over m of flat-reshaped ALLF rows of length 256
__global__ void colmean_k(const float* __restrict__ X, float* __restrict__ mu, int M) {
  int c = blockIdx.x;   // 0..255
  float s = 0.f;
  for (int m = threadIdx.x; m < M; m += blockDim.x) s += X[(size_t)m * 256 + c];
  __shared__ float sh[128];
  sh[threadIdx.x] = s;
  __syncthreads();
  for (int o = 64; o > 0; o >>= 1) {
    if ((int)threadIdx.x < o) sh[threadIdx.x] += sh[threadIdx.x + o];
    __syncthreads();
  }
  if (threadIdx.x == 0) mu[c] = sh[0] / M;
}

__global__ void wv_k(const float* __restrict__ mu, const float* __restrict__ w,
                     const float* __restrict__ b, float* __restrict__ WV) {
  if (threadIdx.x != 0 || blockIdx.x != 0) return;
  float lg[4];
  for (int o = 0; o < 4; ++o) {
    float s = b[o];
    for (int c = 0; c < 256; ++c) s += mu[c] * w[c * 4 + o];
    lg[o] = s;
  }
  float mx = fmaxf(fmaxf(lg[0], lg[1]), fmaxf(lg[2], lg[3]));
  float s = 0.f;
  for (int o = 0; o < 4; ++o) { lg[o] = expf(lg[o] - mx); s += lg[o]; }
  for (int o = 0; o < 4; ++o) WV[o] = lg[o] / s;
}

__global__ void wsum_k(const float* __restrict__ ALLF, const float* __restrict__ WV,
                       float* __restrict__ R, int M) {
  int t = blockIdx.x * blockDim.x + threadIdx.x;
  if (t >= M * DIM) return;
  size_t MD = (size_t)M * DIM;
  R[t] = WV[0] * ALLF[t] + WV[1] * ALLF[MD + t] + WV[2] * ALLF[2 * MD + t] +
         WV[3] * ALLF[3 * MD + t];
}

// ---------------------------------------------------------------------------
extern "C" void kernel_launch(void* const* d_in, const int* in_sizes, int n_in_cnt,
                              void* d_out, int out_size, void* d_ws, size_t ws_size,
                              hipStream_t stream) {
  (void)n_in_cnt; (void)out_size; (void)ws_size;
  const int NI[3] = {128 * 706, 128 * 353, 128 * 177};   // nodes in per layer
  const int NO[3] = {128 * 353, 128 * 177, 128 * 89};    // nodes out per layer
  const int M = 128 * 89;                                 // 11392 final nodes

  const float* x0 = (const float*)d_in[15];
  const int* CL[3] = {(const int*)d_in[12], (const int*)d_in[13], (const int*)d_in[14]};
  const float* Wm[3] = {(const float*)d_in[16], (const float*)d_in[20], (const float*)d_in[24]};
  const float* As[3] = {(const float*)d_in[17], (const float*)d_in[21], (const float*)d_in[25]};
  const float* Ad[3] = {(const float*)d_in[18], (const float*)d_in[22], (const float*)d_in[26]};
  const float* Bi[3] = {(const float*)d_in[19], (const float*)d_in[23], (const float*)d_in[27]};
  const float* w_attn  = (const float*)d_in[28];
  const float* w_lin0  = (const float*)d_in[29];
  const float* b_lin0  = (const float*)d_in[30];
  const float* w_link0 = (const float*)d_in[31];
  const float* w_linkl = (const float*)d_in[32];
  const float* w_link_ = (const float*)d_in[33];

  size_t Emax = 0;
  for (int k = 0; k < 12; ++k) {
    size_t e = (size_t)in_sizes[k] / 2;
    if (e > Emax) Emax = e;
  }

  // carve workspace (floats)
  float* wsp = (float*)d_ws;
  auto take = [&](size_t n) { float* q = wsp; wsp += n; return q; };
  float* H    = take((size_t)NI[0] * DIM);
  float* AGG  = take((size_t)NI[0] * DIM);
  float* SS   = take(NI[0]);
  float* SD   = take(NI[0]);
  float* MX   = take(NI[0]);
  float* Z    = take(NI[0]);
  float* P    = take(Emax);
  float* POOL = take((size_t)NO[0] * DIM);
  float* ALLV = take((size_t)4 * M * DIM);
  float* ALLF = take((size_t)4 * M * DIM);
  float* BM   = take((size_t)M * 4);
  float* STAT = take(128);
  float* G    = take(16);
  float* AT   = take(16);
  float* MU   = take(256);
  float* WV   = take(8);
  float* RA   = take((size_t)M * DIM);
  float* RB   = take((size_t)M * DIM);

  const int T = 256;
  for (int v = 0; v < 4; ++v) {
    const float* xc = x0;
    for (int i = 0; i < 3; ++i) {
      const int ni = NI[i], no = NO[i];
      const int* eb = (const int*)d_in[v * 3 + i];
      const int E = in_sizes[v * 3 + i] / 2;
      const int* es = eb;
      const int* ed = eb + E;

      if (i == 0)
        gemm_n64<3, 0><<<dim3(ni / 128), dim3(256), 0, stream>>>(xc, Wm[i], H);
      else
        gemm_n64<64, 0><<<dim3(ni / 128), dim3(256), 0, stream>>>(xc, Wm[i], H);
      scores_k<<<dim3((ni + T - 1) / T), dim3(T), 0, stream>>>(H, As[i], Ad[i], SS, SD, ni);
      init_agg_k<<<dim3((ni * DIM + T - 1) / T), dim3(T), 0, stream>>>(AGG, MX, Z, ni);
      edge1_k<<<dim3((E + T - 1) / T), dim3(T), 0, stream>>>(es, ed, SS, SD, P, MX, E);
      edge2_k<<<dim3((E + T - 1) / T), dim3(T), 0, stream>>>(ed, P, MX, Z, E);
      edge3_k<<<dim3((E * 16 + T - 1) / T), dim3(T), 0, stream>>>(es, ed, P, Z, H, AGG, E);
      fill_k<<<dim3((no * DIM + T - 1) / T), dim3(T), 0, stream>>>(POOL, 0.f, no * DIM);
      pool_k<<<dim3((ni * DIM + T - 1) / T), dim3(T), 0, stream>>>(AGG, Bi[i], CL[i], POOL, ni);
      bn_stats_k<<<dim3(64), dim3(256), 0, stream>>>(POOL, STAT, no);
      float* dst = (i == 2) ? (ALLV + (size_t)v * M * DIM) : POOL;
      bn_apply_k<<<dim3((no * DIM + T - 1) / T), dim3(T), 0, stream>>>(POOL, STAT, dst, no);

      xc = POOL;
    }
  }

  // view fusion
  bm_k<<<dim3((M + T - 1) / T), dim3(T), 0, stream>>>(ALLV, w_attn, BM, M);
  gram16_k<<<dim3(1), dim3(256), 0, stream>>>(BM, G, M);
  attn4_k<<<dim3(1), dim3(32), 0, stream>>>(G, AT);
  fuse_k<<<dim3((M * DIM + T - 1) / T), dim3(T), 0, stream>>>(ALLV, AT, ALLF, M);
  colmean_k<<<dim3(256), dim3(128), 0, stream>>>(ALLF, MU, M);
  wv_k<<<dim3(1), dim3(32), 0, stream>>>(MU, w_lin0, b_lin0, WV);
  wsum_k<<<dim3((M * DIM + T - 1) / T), dim3(T), 0, stream>>>(ALLF, WV, RA, M);

  // link layers (WMMA GEMMs, selu epilogues)
  gemm_n64<64, 1><<<dim3(M / 128), dim3(256), 0, stream>>>(RA, w_link0, RB);
  gemm_n64<64, 1><<<dim3(M / 128), dim3(256), 0, stream>>>(RB, w_linkl, RA);
  gemm_n64<64, 0><<<dim3(M / 128), dim3(256), 0, stream>>>(RA, w_link_, (float*)d_out);
}